// Geospatial_SelfAttention_24979529793896
// MI455X (gfx1250) — compile-verified
//
#include <hip/hip_runtime.h>
#include <math.h>
#include <stdint.h>

typedef __attribute__((ext_vector_type(2))) float v2f;
typedef __attribute__((ext_vector_type(8))) float v8f;

constexpr int DIM  = 128;  // model dim
constexpr int NH   = 8;    // heads
constexpr int HD   = 16;   // head dim
constexpr int KNN  = 32;   // keys per query

// ---------------------------------------------------------------------------
// Async global -> LDS DMA (CDNA5). lds_off = LDS byte offset (VDST),
// gaddr = 64-bit global address (VADDR). Tracked by ASYNCcnt.
// ---------------------------------------------------------------------------
__device__ inline void async_load_lds_b128(uint32_t lds_off, const void* gaddr) {
  asm volatile("global_load_async_to_lds_b128 %0, %1, off"
               :: "v"(lds_off), "v"(gaddr) : "memory");
}
__device__ inline void wait_asynccnt0() {
  asm volatile("s_wait_asynccnt 0x0" ::: "memory");
}

// ---------------------------------------------------------------------------
// Kernel 1: fused QKV projection. One wave per 16x16 output tile; the A tile
// (rows of x) is loaded once and fed into three f32 WMMA accumulators.
// Y = X @ W^T ; A[m][c] = X[m0+m][c], B[c][n] = W[n0+n][c] (same load pattern).
// ---------------------------------------------------------------------------
__global__ __launch_bounds__(32)
void qkv_proj_kernel(const float* __restrict__ x,
                     const float* __restrict__ Wq,
                     const float* __restrict__ Wk,
                     const float* __restrict__ Wv,
                     float* __restrict__ q,
                     float* __restrict__ k,
                     float* __restrict__ v) {
  const int lane = threadIdx.x;       // 0..31
  const int mr   = lane & 15;         // M index for A, N index for B
  const int kh   = lane >> 4;         // which K-half (0/1)
  const size_t m0 = (size_t)blockIdx.x * 16;
  const int    n0 = blockIdx.y * 16;

  const float* xrow = x  + (m0 + mr) * DIM;
  const float* wqr  = Wq + (size_t)(n0 + mr) * DIM;
  const float* wkr  = Wk + (size_t)(n0 + mr) * DIM;
  const float* wvr  = Wv + (size_t)(n0 + mr) * DIM;

  v8f accq = {}, acck = {}, accv = {};
  for (int c0 = 0; c0 < DIM; c0 += 4) {
    const int co = c0 + 2 * kh;
    v2f a  = *(const v2f*)(xrow + co);
    v2f bq = *(const v2f*)(wqr  + co);
    v2f bk = *(const v2f*)(wkr  + co);
    v2f bv = *(const v2f*)(wvr  + co);
    accq = __builtin_amdgcn_wmma_f32_16x16x4_f32(false, a, false, bq, (short)0, accq, false, false);
    acck = __builtin_amdgcn_wmma_f32_16x16x4_f32(false, a, false, bk, (short)0, acck, false, false);
    accv = __builtin_amdgcn_wmma_f32_16x16x4_f32(false, a, false, bv, (short)0, accv, false, false);
  }

#pragma unroll
  for (int r = 0; r < 8; ++r) {
    const size_t off = (m0 + r + 8 * kh) * DIM + n0 + mr;
    q[off] = accq[r];
    k[off] = acck[r];
    v[off] = accv[r];
  }
}

// ---------------------------------------------------------------------------
// Kernel 2: gather + per-head softmax attention over 32 keys.
// One 256-thread block per (b,l). k/v are L2-resident (10 MB each); the
// 32-row gather is staged into LDS with async global->LDS B128 DMA.
// Wave w (lanes = keys) handles head w; softmax via wave32 shuffles.
// ---------------------------------------------------------------------------
__global__ __launch_bounds__(256)
void attn_kernel(const float* __restrict__ q,
                 const float* __restrict__ k,
                 const float* __restrict__ v,
                 const int*   __restrict__ key_indices,
                 float* __restrict__ attn,
                 int L) {
  __shared__ __align__(16) float qs[DIM];
  __shared__ __align__(16) float ksm[KNN][DIM];
  __shared__ __align__(16) float vsm[KNN][DIM];
  __shared__ int   idxs[KNN];
  __shared__ float Ps[NH][KNN];

  const int tid = threadIdx.x;
  const int bl  = blockIdx.x;          // flattened b*L + l
  const int b   = bl / L;
  const int l   = bl - b * L;

  if (tid < KNN) idxs[tid] = key_indices[(size_t)l * KNN + tid];
  if (tid < DIM) qs[tid]   = q[(size_t)bl * DIM + tid];
  __syncthreads();

  // Gather 32 k-rows and 32 v-rows (128 floats each) into LDS via async DMA.
  // 1024 float4 groups per matrix; 4 iterations of 256 threads.
  for (int g = tid; g < (KNN * DIM) / 4; g += 256) {
    const int j  = g >> 5;             // key slot (32 float4-groups per row)
    const int c4 = (g & 31) * 4;       // float offset within row
    const size_t src = ((size_t)b * L + idxs[j]) * DIM + c4;
    async_load_lds_b128((uint32_t)(uintptr_t)&ksm[j][c4], k + src);
    async_load_lds_b128((uint32_t)(uintptr_t)&vsm[j][c4], v + src);
  }
  wait_asynccnt0();
  __syncthreads();

  // scores: one thread per (head, key); one wave32 == one head
  const int h  = tid >> 5;             // 0..7
  const int kk = tid & 31;             // 0..31
  float s = 0.f;
  const float* qh = qs + h * HD;
  const float* kr = &ksm[kk][h * HD];
#pragma unroll
  for (int dd = 0; dd < HD; ++dd) s += qh[dd] * kr[dd];
  s *= 0.25f;                          // 1/sqrt(16)

  // wave32 softmax
  float m = s;
#pragma unroll
  for (int off = 16; off > 0; off >>= 1) m = fmaxf(m, __shfl_xor(m, off, 32));
  float p = __expf(s - m);
  float sum = p;
#pragma unroll
  for (int off = 16; off > 0; off >>= 1) sum += __shfl_xor(sum, off, 32);
  Ps[h][kk] = p / sum;
  __syncthreads();

  // weighted sum of values: one thread per output channel
  if (tid < DIM) {
    const int hh = tid >> 4;
    const int dd = tid & 15;
    float o = 0.f;
#pragma unroll
    for (int j = 0; j < KNN; ++j) o += Ps[hh][j] * vsm[j][hh * HD + dd];
    attn[(size_t)bl * DIM + tid] = o;
  }
}

// ---------------------------------------------------------------------------
// Kernel 3: output projection with bias: Y = attn @ Wo^T + bo
// ---------------------------------------------------------------------------
__global__ __launch_bounds__(32)
void out_proj_kernel(const float* __restrict__ a,
                     const float* __restrict__ Wo,
                     const float* __restrict__ bo,
                     float* __restrict__ y) {
  const int lane = threadIdx.x;
  const int mr   = lane & 15;
  const int kh   = lane >> 4;
  const size_t m0 = (size_t)blockIdx.x * 16;
  const int    n0 = blockIdx.y * 16;

  const float* arow = a  + (m0 + mr) * DIM;
  const float* wrow = Wo + (size_t)(n0 + mr) * DIM;

  v8f acc = {};
  for (int c0 = 0; c0 < DIM; c0 += 4) {
    const int co = c0 + 2 * kh;
    v2f av = *(const v2f*)(arow + co);
    v2f bv = *(const v2f*)(wrow + co);
    acc = __builtin_amdgcn_wmma_f32_16x16x4_f32(false, av, false, bv, (short)0, acc, false, false);
  }

  const float bias = bo[n0 + mr];
#pragma unroll
  for (int r = 0; r < 8; ++r) {
    y[(m0 + r + 8 * kh) * DIM + n0 + mr] = acc[r] + bias;
  }
}

// ---------------------------------------------------------------------------
extern "C" void kernel_launch(void* const* d_in, const int* in_sizes, int n_in,
                              void* d_out, int out_size, void* d_ws, size_t ws_size,
                              hipStream_t stream) {
  (void)n_in; (void)out_size; (void)ws_size;

  const float* x    = (const float*)d_in[0];
  const int*   kidx = (const int*)  d_in[1];
  const float* Wq   = (const float*)d_in[2];
  const float* Wk   = (const float*)d_in[3];
  const float* Wv   = (const float*)d_in[4];
  const float* Wo   = (const float*)d_in[5];
  const float* bo   = (const float*)d_in[6];
  float* out = (float*)d_out;

  const int    L = in_sizes[1] / KNN;                 // 5000
  const size_t M = (size_t)in_sizes[0] / DIM;         // B*L = 20000

  float* q    = (float*)d_ws;
  float* kmat = q    + M * DIM;
  float* vmat = kmat + M * DIM;
  float* attn = vmat + M * DIM;

  dim3 gridG((unsigned)(M / 16), DIM / 16);           // 1250 x 8 tiles
  qkv_proj_kernel<<<gridG, 32, 0, stream>>>(x, Wq, Wk, Wv, q, kmat, vmat);
  attn_kernel<<<dim3((unsigned)M), 256, 0, stream>>>(q, kmat, vmat, kidx, attn, L);
  out_proj_kernel<<<gridG, 32, 0, stream>>>(attn, Wo, bo, out);
}